// Cosine_Similarity_28621662061011
// MI455X (gfx1250) — compile-verified
//
#include <hip/hip_runtime.h>

typedef float v2f __attribute__((ext_vector_type(2)));
typedef float v8f __attribute__((ext_vector_type(8)));
typedef int   v4i __attribute__((ext_vector_type(4)));

#define HH   2048
#define WW   2048
#define CC   16
#define TILE 32
#define HT   (TILE + 2)   /* 34: tile + 1-pixel halo each side */
#define EPSV 1e-8f

// ---- async global->LDS copy support (gfx1250 GLOBAL_LOAD_ASYNC_TO_LDS_B128) ----
#if defined(__has_builtin)
#if __has_builtin(__builtin_amdgcn_global_load_async_to_lds_b128)
#define ASYNC_LDS_MODE 2
#elif __has_builtin(__builtin_amdgcn_global_load_lds)
#define ASYNC_LDS_MODE 1
#endif
#endif
#ifndef ASYNC_LDS_MODE
#define ASYNC_LDS_MODE 0
#endif

typedef __attribute__((address_space(1))) v4i as1_v4i;   // prints as "int4 __device__ *"
typedef __attribute__((address_space(3))) v4i as3_v4i;

__device__ __forceinline__ void async_copy16(const void* gsrc, void* ldst) {
#if ASYNC_LDS_MODE == 2
  __builtin_amdgcn_global_load_async_to_lds_b128(
      (as1_v4i*)(void*)gsrc, (as3_v4i*)ldst, /*offset=*/0, /*cpol=*/0);
#elif ASYNC_LDS_MODE == 1
  __builtin_amdgcn_global_load_lds(
      (as1_v4i*)(void*)gsrc, (as3_v4i*)ldst, /*size=*/16, /*offset=*/0, /*aux=*/0);
#else
  *reinterpret_cast<float4*>(ldst) = *reinterpret_cast<const float4*>(gsrc);
#endif
}

__device__ __forceinline__ void async_drain() {
#if ASYNC_LDS_MODE == 2
#if __has_builtin(__builtin_amdgcn_s_wait_asynccnt)
  __builtin_amdgcn_s_wait_asynccnt(0);
#else
  asm volatile("s_wait_asynccnt 0x0" ::: "memory");
#endif
#elif ASYNC_LDS_MODE == 1
  asm volatile("s_wait_asynccnt 0x0" ::: "memory");
#endif
}

__global__ __launch_bounds__(256) void cosine_sim_wmma(const float* __restrict__ g,
                                                       float* __restrict__ out) {
  __shared__ float s_img[HT * HT * CC];   // 73,984 B
  __shared__ float s_nrm[HT * HT];        //  4,624 B

  const int tid = threadIdx.x;
  const int ty0 = blockIdx.y * TILE;
  const int tx0 = blockIdx.x * TILE;

  // ---- Stage 1: DMA haloed tile into LDS (zero-pad out-of-bounds) ----
  const float4* g4 = reinterpret_cast<const float4*>(g);
  for (int i = tid; i < HT * HT * 4; i += 256) {
    const int pix = i >> 2;
    const int ch  = i & 3;
    const int ly  = pix / HT;
    const int lx  = pix - ly * HT;
    const int gy  = ty0 - 1 + ly;
    const int gx  = tx0 - 1 + lx;
    float* ldst = &s_img[pix * CC + ch * 4];
    if (gy >= 0 && gy < HH && gx >= 0 && gx < WW) {
      async_copy16(g4 + (gy * WW + gx) * 4 + ch, ldst);   // EXEC-masked per lane
    } else {
      *reinterpret_cast<float4*>(ldst) = make_float4(0.f, 0.f, 0.f, 0.f);
    }
  }
  async_drain();       // each wave drains its ASYNCcnt before signaling barrier
  __syncthreads();

  // ---- Stage 2: per-pixel L2 norms (zero-padded pixels give n = 0) ----
  for (int p = tid; p < HT * HT; p += 256) {
    const float* v = &s_img[p * CC];
    float s = 0.f;
#pragma unroll
    for (int c = 0; c < CC; ++c) s += v[c] * v[c];
    s_nrm[p] = sqrtf(s);
  }
  __syncthreads();

  // ---- Stage 3: Gram-diagonal via V_WMMA_F32_16X16X4_F32 ----
  const int wave = tid >> 5;
  const int lane = tid & 31;
  const int Mn   = lane & 15;          // M (for A) == N (for B)
  const int h2   = (lane >> 4) << 1;   // K sub-offset: 0 (lanes 0-15) / 2 (lanes 16-31)

  const int dyv[4] = {-1, -1, 1, 1};
  const int dxv[4] = {-1, 1, -1, 1};

  for (int it = 0; it < 8; ++it) {
    const int seg = wave * 8 + it;          // 64 segments of 16 pixels
    const int ly  = (seg >> 1) + 1;         // center row in halo coords (1..32)
    const int lx0 = (seg & 1) * 16 + 1;     // center col base (1 or 17)

    v8f acc[4] = {v8f{}, v8f{}, v8f{}, v8f{}};

#pragma unroll
    for (int kc = 0; kc < CC; kc += 4) {
      const int c0 = kc + h2;
      // A fragment: center vectors, one float2 (channels c0, c0+1) per lane
      v2f a = *reinterpret_cast<const v2f*>(
          &s_img[(ly * HT + lx0 + Mn) * CC + c0]);
#pragma unroll
      for (int d = 0; d < 4; ++d) {
        v2f b = *reinterpret_cast<const v2f*>(
            &s_img[((ly + dyv[d]) * HT + (lx0 + dxv[d] + Mn)) * CC + c0]);
        acc[d] = __builtin_amdgcn_wmma_f32_16x16x4_f32(
            /*neg_a=*/false, a, /*neg_b=*/false, b,
            /*c_mod=*/(short)0, acc[d], /*reuse_a=*/false, /*reuse_b=*/false);
      }
    }

    // Extract Gram diagonals: pixel p's dot lives at VGPR (p&7),
    // lane p (p<8) or lane p+16 (p>=8) -> select into lanes {0..7, 24..31}.
    float diag[4];
#pragma unroll
    for (int d = 0; d < 4; ++d) {
      float v = 0.f;
#pragma unroll
      for (int r = 0; r < 8; ++r)
        v = (lane == r || lane == r + 24) ? acc[d][r] : v;
      diag[d] = v;
    }

    const bool active = (lane < 8) || (lane >= 24);
    const int  p      = (lane < 8) ? lane : lane - 16;   // pixel index 0..15

    const float nc = s_nrm[ly * HT + lx0 + p];
    float sim = 0.f;
#pragma unroll
    for (int d = 0; d < 4; ++d) {
      const float nd = s_nrm[(ly + dyv[d]) * HT + (lx0 + dxv[d] + p)];
      sim += diag[d] / fmaxf(nc * nd, EPSV);
    }

    if (active)
      out[(ty0 + ly - 1) * WW + (tx0 + lx0 - 1 + p)] = sim;
  }
}

extern "C" void kernel_launch(void* const* d_in, const int* in_sizes, int n_in,
                              void* d_out, int out_size, void* d_ws, size_t ws_size,
                              hipStream_t stream) {
  (void)in_sizes; (void)n_in; (void)d_ws; (void)ws_size; (void)out_size;
  const float* img = (const float*)d_in[0];
  float* out = (float*)d_out;
  dim3 grid(WW / TILE, HH / TILE);   // 64 x 64
  dim3 block(256);
  cosine_sim_wmma<<<grid, block, 0, stream>>>(img, out);
}